// WrapperLinear_33122787787467
// MI455X (gfx1250) — compile-verified
//
#include <hip/hip_runtime.h>

typedef __attribute__((ext_vector_type(2))) float v2f;
typedef __attribute__((ext_vector_type(4))) float v4f;
typedef __attribute__((ext_vector_type(8))) float v8f;

#define IN_F   4096
#define OUT_F  4096
#define M_DIM  32
#define GROUP  128
#define MAXQF  15.0f
#define GROUPS_PER_ROW (IN_F / GROUP)   // 32
#define ROWS_PER_WG    16               // N-tile width

__global__ __launch_bounds__(256, 2)
void wrapper_linear_fused_kernel(const float* __restrict__ x,
                                 const float* __restrict__ weight,
                                 const float* __restrict__ bias,
                                 const float* __restrict__ value,
                                 const float* __restrict__ min_scale,
                                 const float* __restrict__ max_scale,
                                 float* __restrict__ out)
{
    __shared__ float sScale[ROWS_PER_WG][GROUPS_PER_ROW];   // 2 KB
    __shared__ float sInvS [ROWS_PER_WG][GROUPS_PER_ROW];   // 2 KB
    __shared__ float sZp   [ROWS_PER_WG][GROUPS_PER_ROW];   // 2 KB
    __shared__ float sRed[8][32][8];                        // 8 KB partial C

    const int tid  = threadIdx.x;
    const int lane = tid & 31;
    const int wave = tid >> 5;
    const int n0   = blockIdx.x * ROWS_PER_WG;

    // ---------------- Phase A: per-group scale / inv-scale / zero-point ----------------
    // 16 rows x 32 groups = 512 groups per WG; each wave handles 64 groups,
    // one full group (128 floats) per iteration: 32 lanes x float4.
    for (int i = 0; i < 64; ++i) {
        const int gid = wave * 64 + i;
        const int r   = gid >> 5;          // row within tile
        const int g   = gid & 31;          // group within row
        const int n   = n0 + r;
        const float* wp = weight + (size_t)n * IN_F + g * GROUP + lane * 4;
        const v4f w4 = *(const v4f*)wp;
        float mn = fminf(fminf(w4.x, w4.y), fminf(w4.z, w4.w));
        float mx = fmaxf(fmaxf(w4.x, w4.y), fmaxf(w4.z, w4.w));
        #pragma unroll
        for (int m = 16; m >= 1; m >>= 1) {
            mn = fminf(mn, __shfl_xor(mn, m, 32));
            mx = fmaxf(mx, __shfl_xor(mx, m, 32));
        }
        if (lane == 0) {
            const int gidx = n * GROUPS_PER_ROW + g;
            const float ms = fminf(fmaxf(min_scale[gidx], -1.0f), 0.0f);
            const float xs = fminf(fmaxf(max_scale[gidx], -1.0f), 0.0f);
            const float wmin_t = fminf(mn, 0.0f) * (ms + 1.0f);
            const float wmax_t = fmaxf(mx, 0.0f) * (xs + 1.0f);
            float wmax = fmaxf(wmax_t, wmin_t);
            float wmin = fminf(wmax_t, wmin_t);
            if (wmin == 0.0f && wmax == 0.0f) { wmin = -1.0f; wmax = 1.0f; }
            const float scale = (wmax - wmin) / MAXQF;
            const float zp    = rintf(-wmin / scale);   // round half-to-even
            sScale[r][g] = scale;
            sInvS [r][g] = 1.0f / scale;                // exact divide, once per group
            sZp   [r][g] = zp;
        }
    }
    __syncthreads();

    // ---------------- Phase B: dequant-on-the-fly fp32 WMMA GEMM ----------------
    // 8 waves = 2 M-tiles (rows 0-15 / 16-31 of x) x 4 K-partitions of 1024.
    // Outer loop over the 8 quant groups in this wave's K range so scale/zp/inv
    // live in VGPRs across the 32-step inner loop (no per-step LDS traffic).
    const int mtile = wave >> 2;       // which 16 rows of x
    const int ksub  = wave & 3;        // K partition
    const int hl    = lane >> 4;       // lane half selects K pair {0,1} vs {2,3}
    const int l16   = lane & 15;
    const int mrow  = mtile * 16 + l16;
    const int nrow  = n0 + l16;

    const float* xp = x      + (size_t)mrow * IN_F;
    const float* wp = weight + (size_t)nrow * IN_F;
    const float* gp = value  + (size_t)nrow * IN_F;

    v8f c = {0.0f, 0.0f, 0.0f, 0.0f, 0.0f, 0.0f, 0.0f, 0.0f};

    const int grpBeg = ksub * (GROUPS_PER_ROW / 4);   // 8 groups per wave

    for (int grp = grpBeg; grp < grpBeg + 8; ++grp) {
        const float s     = sScale[l16][grp];
        const float inv_s = sInvS [l16][grp];
        const float z     = sZp   [l16][grp];
        const float sz    = s * z;                    // wq = s*q - s*z
        const int   kbase = grp * GROUP;

        #pragma unroll 4
        for (int k = kbase; k < kbase + GROUP; k += 4) {
            const int kk = k + 2 * hl;
            // keep the weight/value streams ~2KB ahead of the loads
            __builtin_prefetch(wp + kk + 512, 0, 1);
            __builtin_prefetch(gp + kk + 512, 0, 1);

            const v2f a  = *(const v2f*)(xp + kk);    // A: 16x4 fp32 fragment
            const v2f w  = *(const v2f*)(wp + kk);
            const v2f gv = *(const v2f*)(gp + kk);

            v2f b;                                    // B: 4x16 fp32 fragment (wq)
            {
                const float q0 = fminf(fmaxf(rintf(__builtin_fmaf(w.x, inv_s, gv.x)) + z, 0.0f), MAXQF);
                const float q1 = fminf(fmaxf(rintf(__builtin_fmaf(w.y, inv_s, gv.y)) + z, 0.0f), MAXQF);
                b.x = __builtin_fmaf(s, q0, -sz);
                b.y = __builtin_fmaf(s, q1, -sz);
            }

            c = __builtin_amdgcn_wmma_f32_16x16x4_f32(
                    /*neg_a=*/false, a, /*neg_b=*/false, b,
                    /*c_mod=*/(short)0, c, /*reuse_a=*/false, /*reuse_b=*/false);
        }
    }

    // ---------------- Phase C: cross-wave reduction + bias + store ----------------
    #pragma unroll
    for (int j = 0; j < 8; ++j) sRed[wave][lane][j] = c[j];
    __syncthreads();

    // 512 output elements (32 x 16 tile); 256 threads -> 2 each.
    for (int e = tid; e < 512; e += 256) {
        const int mt  = e >> 8;          // M-tile
        const int idx = e & 255;
        const int vg  = idx >> 5;        // C VGPR index 0..7
        const int ln  = idx & 31;        // lane within wave
        float sum = 0.0f;
        #pragma unroll
        for (int w2 = 0; w2 < 4; ++w2) sum += sRed[mt * 4 + w2][ln][vg];
        // C layout: VGPR j -> M=j (lanes 0-15) / M=8+j (lanes 16-31); N = lane%16
        const int Np = ln & 15;
        const int Mp = (ln >> 4) * 8 + vg;
        const int m  = mt * 16 + Mp;
        const int n  = n0 + Np;
        out[(size_t)m * OUT_F + n] = sum + bias[n];
    }
}

extern "C" void kernel_launch(void* const* d_in, const int* in_sizes, int n_in,
                              void* d_out, int out_size, void* d_ws, size_t ws_size,
                              hipStream_t stream) {
    (void)in_sizes; (void)n_in; (void)d_ws; (void)ws_size; (void)out_size;
    const float* x         = (const float*)d_in[0];
    const float* weight    = (const float*)d_in[1];
    const float* bias      = (const float*)d_in[2];
    const float* value     = (const float*)d_in[3];
    const float* min_scale = (const float*)d_in[4];
    const float* max_scale = (const float*)d_in[5];
    float* out = (float*)d_out;

    dim3 grid(OUT_F / ROWS_PER_WG);   // 256 workgroups
    dim3 block(256);                  // 8 wave32s
    wrapper_linear_fused_kernel<<<grid, block, 0, stream>>>(
        x, weight, bias, value, min_scale, max_scale, out);
}